// LocalAttentionBlock_48885317763624
// MI455X (gfx1250) — compile-verified
//
#include <hip/hip_runtime.h>

typedef __attribute__((ext_vector_type(2))) float v2f;
typedef __attribute__((ext_vector_type(8))) float v8f;

#define C_   64
#define DIM  24
#define NVOX (DIM * DIM * DIM)   // 13824
#define BATCH 2
#define HEADS 4
#define HD    16                 // C_/HEADS
#define NTILES (NVOX / 16)       // 864
#define NCHUNKS 27               // 864 / (8 waves * 4 tiles)

// Compute Q^T/K^T/V^T = W * X + bias, channel-major output (B, C, N).
// A = W (M=out-channel, K=in-channel), B = X[b] (K=channel rows, N=voxel cols).
// Per wave: one 16-row M tile, 4 N tiles of 16 voxels, K=64 in 16 WMMA steps.
__global__ __launch_bounds__(256) void qkv_gemm_wmma(
    const float* __restrict__ x,
    const float* __restrict__ Wq, const float* __restrict__ bq,
    const float* __restrict__ Wk, const float* __restrict__ bk,
    const float* __restrict__ Wv, const float* __restrict__ bv,
    float* __restrict__ qkvT)
{
    int blk = blockIdx.x;
    int nchunk = blk % NCHUNKS; blk /= NCHUNKS;
    int mtile  = blk % 4;       blk /= 4;
    int mat    = blk % 3;
    int b      = blk / 3;

    const float* W    = (mat == 0) ? Wq : (mat == 1) ? Wk : Wv;
    const float* bias = (mat == 0) ? bq : (mat == 1) ? bk : bv;
    float* dst = qkvT + (size_t)mat * BATCH * C_ * NVOX + (size_t)b * C_ * NVOX;

    const int wave  = threadIdx.x >> 5;
    const int lane  = threadIdx.x & 31;
    const int lhalf = lane >> 4;   // 0 or 1 (upper K/M half)
    const int l15   = lane & 15;

    // A fragments (16x4 f32 layout): lane = M row, lhalf selects K pair.
    // W row-major -> the two K values are contiguous (single b64 load each).
    v2f afrag[16];
    const int m_a = mtile * 16 + l15;
    #pragma unroll
    for (int ks = 0; ks < 16; ++ks) {
        const int ka = ks * 4 + lhalf * 2;
        afrag[ks].x = W[m_a * C_ + ka];
        afrag[ks].y = W[m_a * C_ + ka + 1];
    }

    // Bias broadcast into the 16x16 f32 C/D layout (VGPR i -> row i + 8*lhalf).
    v8f cinit;
    #pragma unroll
    for (int i = 0; i < 8; ++i)
        cinit[i] = bias[mtile * 16 + i + 8 * lhalf];

    const float* Xb = x + (size_t)b * C_ * NVOX;

    #pragma unroll
    for (int t = 0; t < 4; ++t) {
        const int ntile = nchunk * 32 + wave * 4 + t;
        const int n = ntile * 16 + l15;       // voxel column (lane-contiguous)
        v8f acc = cinit;
        #pragma unroll
        for (int ks = 0; ks < 16; ++ks) {
            // B fragment (4x16 f32): lane = N col, lhalf selects K pair.
            // X is channel-major -> each row load is coalesced across lanes.
            const int kb = ks * 4 + lhalf * 2;
            v2f bfrag;
            bfrag.x = Xb[(size_t)kb * NVOX + n];
            bfrag.y = Xb[(size_t)(kb + 1) * NVOX + n];
            acc = __builtin_amdgcn_wmma_f32_16x16x4_f32(
                false, afrag[ks], false, bfrag, (short)0, acc, false, false);
        }
        // D tile is (out-channel row, voxel col) -> store channel-major directly.
        float* dptr = dst + n;
        #pragma unroll
        for (int i = 0; i < 8; ++i) {
            const int row = mtile * 16 + i + 8 * lhalf;
            dptr[(size_t)row * NVOX] = acc[i];
        }
    }
}

// One thread per (batch, head, voxel). Lanes map to consecutive voxels so all
// q/k/v/x accesses are coalesced. OOB neighbors: score=0 kept in softmax
// denominator (zero-padded k), zero v contribution — matches the reference.
__global__ __launch_bounds__(256) void local_attn_kernel(
    const float* __restrict__ x,
    const float* __restrict__ qkvT,
    float* __restrict__ out)
{
    const float* qT = qkvT;
    const float* kT = qkvT + (size_t)BATCH * C_ * NVOX;
    const float* vT = qkvT + (size_t)2 * BATCH * C_ * NVOX;

    const int b    = blockIdx.y;
    const int head = threadIdx.x >> 6;                    // 4 heads
    const int n    = blockIdx.x * 64 + (threadIdx.x & 63);// 64 voxels/block
    const int d    = n / (DIM * DIM);
    const int rem  = n - d * DIM * DIM;
    const int h    = rem / DIM;
    const int w    = rem - h * DIM;
    const int c0   = head * HD;

    const size_t base = (size_t)(b * C_ + c0) * NVOX;

    float qv[HD];
    #pragma unroll
    for (int cc = 0; cc < HD; ++cc)
        qv[cc] = qT[base + (size_t)cc * NVOX + n] * 0.25f;  // hd^-0.5 = 0.25

    int   nbr[27];
    float scores[27];
    #pragma unroll
    for (int idx = 0; idx < 27; ++idx) {
        const int i = idx / 9, j = (idx / 3) % 3, l = idx % 3;
        const int dd = d + i - 1, hh = h + j - 1, ww = w + l - 1;
        const bool ok = ((unsigned)dd < (unsigned)DIM) &&
                        ((unsigned)hh < (unsigned)DIM) &&
                        ((unsigned)ww < (unsigned)DIM);
        const int nn = (dd * DIM + hh) * DIM + ww;
        nbr[idx] = ok ? nn : -1;
        float s = 0.0f;
        if (ok) {
            const float* kp = kT + base + nn;
            #pragma unroll
            for (int cc = 0; cc < HD; ++cc)
                s = fmaf(qv[cc], kp[(size_t)cc * NVOX], s);
        }
        scores[idx] = s;
    }

    float mx = scores[0];
    #pragma unroll
    for (int idx = 1; idx < 27; ++idx) mx = fmaxf(mx, scores[idx]);
    float ssum = 0.0f;
    #pragma unroll
    for (int idx = 0; idx < 27; ++idx) {
        scores[idx] = __expf(scores[idx] - mx);
        ssum += scores[idx];                  // OOB entries included (exp(-mx))
    }
    const float inv = 1.0f / ssum;

    float acc[HD] = {};
    #pragma unroll
    for (int idx = 0; idx < 27; ++idx) {
        if (nbr[idx] >= 0) {
            const float p = scores[idx] * inv;
            const float* vp = vT + base + nbr[idx];
            #pragma unroll
            for (int cc = 0; cc < HD; ++cc)
                acc[cc] = fmaf(p, vp[(size_t)cc * NVOX], acc[cc]);
        }
    }

    // Residual + store, already in (B, C, D, H, W) order.
    #pragma unroll
    for (int cc = 0; cc < HD; ++cc)
        out[base + (size_t)cc * NVOX + n] = x[base + (size_t)cc * NVOX + n] + acc[cc];
}

extern "C" void kernel_launch(void* const* d_in, const int* in_sizes, int n_in,
                              void* d_out, int out_size, void* d_ws, size_t ws_size,
                              hipStream_t stream) {
    const float* x  = (const float*)d_in[0];
    // d_in[1] = cemb (unused by the reference forward)
    const float* Wq = (const float*)d_in[2];
    const float* bq = (const float*)d_in[3];
    const float* Wk = (const float*)d_in[4];
    const float* bk = (const float*)d_in[5];
    const float* Wv = (const float*)d_in[6];
    const float* bv = (const float*)d_in[7];
    float* out  = (float*)d_out;
    float* qkvT = (float*)d_ws;   // 3 * B * C * N floats ≈ 21.2 MB

    // 2 batches * 3 matrices * 4 M-tiles * 27 N-chunks = 648 blocks
    qkv_gemm_wmma<<<dim3(BATCH * 3 * 4 * NCHUNKS), dim3(256), 0, stream>>>(
        x, Wq, bq, Wk, bk, Wv, bv, qkvT);

    // 216 voxel-groups * 2 batches, 256 threads = 64 voxels x 4 heads
    local_attn_kernel<<<dim3(NVOX / 64, BATCH), dim3(256), 0, stream>>>(
        x, qkvT, out);
}